// AsterAttentionRecognitionHead_70944269796149
// MI455X (gfx1250) — compile-verified
//
#include <hip/hip_runtime.h>
#include <hip/hip_bf16.h>
#include <math.h>
#include <stdint.h>

// ---------------- problem constants (match reference) ----------------
#define BB     512   // batch
#define TT     64    // encoder seq len
#define DD     512   // input dim
#define HH     256   // hidden dim
#define AA     256   // attention dim
#define CC     96    // num classes
#define NSTEPS 26    // decode steps

#define KC     64    // K-chunk staged in LDS per iteration
#define APAD   68    // padded LDS row stride (68 floats -> conflict-free b64 reads)

// ---------------- CDNA5 WMMA fp32 types ----------------
typedef __attribute__((ext_vector_type(2))) float v2f;
typedef __attribute__((ext_vector_type(8))) float v8f;

// ---------------------------------------------------------------------
// fp32 WMMA GEMM:  Cmat[M,N] = A[M,K] * W[N,K]^T + bias[N]
// Row-major A/W/C.  M,N multiples of 16, K multiple of 64.
//
// Block = 256 threads = 8 wave32.  Block computes one 16(M) x 128(N) strip:
//   - the 16 x KC A-chunk is staged into LDS once per block with
//     GLOBAL_LOAD_ASYNC_TO_LDS_B128 (double-buffered, overlapped with WMMA)
//   - each wave owns one 16x16 N-tile and runs V_WMMA_F32_16X16X4_F32
//     over the chunk (A frags via ds_load_b64, W frags via global b64).
//
// VGPR layouts per CDNA5 ISA 7.12.2:
//   A 16x4 : lanes 0-15 K={0,1}, lanes 16-31 K={2,3}, row = lane&15
//   B 4x16 : lanes 0-15 K={0,1}, lanes 16-31 K={2,3}, col = lane&15
//   C 16x16: VGPR v -> row v (lanes 0-15) / v+8 (lanes 16-31), col = lane&15
// ---------------------------------------------------------------------
__global__ void __launch_bounds__(256)
wmma_gemm_bias_f32(const float* __restrict__ A, int lda,
                   const float* __restrict__ W, int ldw,
                   const float* __restrict__ bias,
                   float* __restrict__ Cmat, int ldc,
                   int M, int N, int K)
{
    __shared__ float s_a[2][16 * APAD];

    const int tid  = threadIdx.x;
    const int lane = tid & 31;
    const int wid  = tid >> 5;                    // 8 waves
    const int tm   = blockIdx.x;                  // M tile
    const int tn   = blockIdx.y * 8 + wid;        // N tile owned by this wave
    const bool active = (tn * 16) < N;            // wave-uniform

    // ---- cooperative async staging: one b128 (16B) per thread per chunk
    const int srow = tid >> 4;                    // 0..15  (A row in tile)
    const int scol = (tid & 15) << 2;             // 0,4,...,60 (col base)
    const unsigned long long gA = (unsigned long long)(uintptr_t)
        (A + (size_t)(tm * 16 + srow) * lda + scol);
    // generic->LDS offset: LDS aperture low 32 bits == LDS byte offset
    const uint32_t ldsDst0 = (uint32_t)(uintptr_t)&s_a[0][srow * APAD + scol];
    const uint32_t ldsDst1 = (uint32_t)(uintptr_t)&s_a[1][srow * APAD + scol];

    // ---- per-lane fragment coordinates
    const int rloc = lane & 15;
    const int koff = (lane >> 4) << 1;            // 0 | 2
    const int colB = tn * 16 + rloc;
    const float* __restrict__ wRow = W + (size_t)colB * ldw + koff;

    v8f acc = {0.f, 0.f, 0.f, 0.f, 0.f, 0.f, 0.f, 0.f};

    const int nch = K >> 6;                       // K / 64

    // prefetch chunk 0 into buffer 0
    asm volatile("global_load_async_to_lds_b128 %0, %1, off"
                 :: "v"(ldsDst0), "v"(gA) : "memory");

    for (int c = 0; c < nch; ++c) {
        if (c + 1 < nch) {
            const uint32_t dst = ((c + 1) & 1) ? ldsDst1 : ldsDst0;
            asm volatile("global_load_async_to_lds_b128 %0, %1, off"
                         :: "v"(dst),
                            "v"(gA + (unsigned long long)(c + 1) * (KC * 4ull))
                         : "memory");
            asm volatile("s_wait_asynccnt 0x1" ::: "memory");  // chunk c done
        } else {
            asm volatile("s_wait_asynccnt 0x0" ::: "memory");  // last chunk done
        }
        __syncthreads();

        if (active) {
            const float* sa = &s_a[c & 1][rloc * APAD + koff];
            const float* wk = wRow + c * KC;
#pragma unroll
            for (int kk = 0; kk < KC; kk += 4) {
                const v2f af = *(const v2f*)(sa + kk);   // ds_load_b64
                const v2f bf = *(const v2f*)(wk + kk);   // global_load_b64
                // (neg_a, A, neg_b, B, c_mod, C, reuse_a, reuse_b)
                acc = __builtin_amdgcn_wmma_f32_16x16x4_f32(
                          false, af, false, bf, (short)0, acc, false, false);
            }
        }
        __syncthreads();   // protect buffer reuse before next prefetch lands
    }

    if (active) {
        const float bv    = bias[colB];
        const int   rbase = tm * 16 + ((lane >> 4) << 3);
#pragma unroll
        for (int v = 0; v < 8; ++v)
            Cmat[(size_t)(rbase + v) * ldc + colB] = acc[v] + bv;
    }
}

// ---------------------------------------------------------------------
// Fused attention step for one batch row b (one 256-thread block):
//   v[t]   = sum_a tanh(sProj[b,a] + xProj[b,t,a]) * Ww[a] + bw
//   alpha  = softmax_t(v)
//   ctx[d] = sum_t alpha[t] * img[b,t,d]
//   u[b]   = [ emb[y_b] (A) | ctx (D) ]     (GRU input, K = A + D = 768)
// ---------------------------------------------------------------------
__global__ void __launch_bounds__(256)
attention_step(const float* __restrict__ xProj,   // [B,T,A]
               const float* __restrict__ sProj,   // [B,A]
               const float* __restrict__ img,     // [B,T,D]
               const float* __restrict__ Ww,      // [A]
               const float* __restrict__ bw,      // [1]
               const float* __restrict__ emb,     // [C+1, A]
               const int*   __restrict__ label,   // [B, NSTEPS]
               float*       __restrict__ u,       // [B, A+D]
               int step)
{
    const int b    = blockIdx.x;
    const int tid  = threadIdx.x;     // 0..255
    const int lane = tid & 31;
    const int wid  = tid >> 5;        // 8 waves

    __shared__ float s_sp[AA];
    __shared__ float s_ww[AA];
    __shared__ float s_v[TT];

    s_sp[tid] = sProj[(size_t)b * AA + tid];
    s_ww[tid] = Ww[tid];
    __syncthreads();

    const float bw0 = bw[0];

    // ---- attention energies: each wave handles 8 of the 64 t positions
    for (int t = wid; t < TT; t += 8) {
        const float* xp = xProj + ((size_t)b * TT + t) * AA;
        float p = 0.f;
#pragma unroll
        for (int a = lane; a < AA; a += 32)
            p += tanhf(s_sp[a] + xp[a]) * s_ww[a];
#pragma unroll
        for (int off = 16; off; off >>= 1)
            p += __shfl_xor(p, off, 32);
        if (lane == 0) s_v[t] = p + bw0;
    }
    __syncthreads();

    // ---- softmax over T=64 by wave 0 (2 values per lane)
    if (tid < 32) {
        float v0 = s_v[tid], v1 = s_v[tid + 32];
        float m = fmaxf(v0, v1);
#pragma unroll
        for (int off = 16; off; off >>= 1)
            m = fmaxf(m, __shfl_xor(m, off, 32));
        float e0 = expf(v0 - m), e1 = expf(v1 - m);
        float s = e0 + e1;
#pragma unroll
        for (int off = 16; off; off >>= 1)
            s += __shfl_xor(s, off, 32);
        float inv = 1.0f / s;
        s_v[tid]      = e0 * inv;
        s_v[tid + 32] = e1 * inv;
    }
    __syncthreads();

    // ---- context vector: ctx[d] = sum_t alpha[t]*img[b,t,d]
    for (int d = tid; d < DD; d += 256) {
        const float* ib = img + (size_t)b * TT * DD + d;
        float acc = 0.f;
#pragma unroll 8
        for (int t = 0; t < TT; ++t)
            acc += s_v[t] * ib[(size_t)t * DD];
        u[(size_t)b * (AA + DD) + AA + d] = acc;
    }

    // ---- embedding part of GRU input (teacher forcing: y=0 at step 0)
    const int y = (step == 0) ? 0 : label[(size_t)b * NSTEPS + step];
    u[(size_t)b * (AA + DD) + tid] = emb[(size_t)y * AA + tid];
}

// ---------------------------------------------------------------------
// GRU gate nonlinearity + hidden-state update (torch gate order r,z,n):
//   r = sig(gi_r+gh_r); z = sig(gi_z+gh_z); n = tanh(gi_n + r*gh_n)
//   h = (1-z)*n + z*h
// ---------------------------------------------------------------------
__global__ void __launch_bounds__(256)
gru_update(const float* __restrict__ gi,   // [B,3H]
           const float* __restrict__ gh,   // [B,3H]
           float*       __restrict__ h)    // [B,H] (in/out)
{
    const int b = blockIdx.x;
    const int j = threadIdx.x;            // 0..255
    const size_t o3 = (size_t)b * (3 * HH);

    const float r = 1.f / (1.f + expf(-(gi[o3 + j]          + gh[o3 + j])));
    const float z = 1.f / (1.f + expf(-(gi[o3 + HH + j]     + gh[o3 + HH + j])));
    const float n = tanhf(gi[o3 + 2 * HH + j] + r * gh[o3 + 2 * HH + j]);

    const size_t oh = (size_t)b * HH + j;
    const float hv = h[oh];
    h[oh] = (1.f - z) * n + z * hv;
}

__global__ void zero_f32(float* __restrict__ p, int n)
{
    int i = blockIdx.x * blockDim.x + threadIdx.x;
    if (i < n) p[i] = 0.f;
}

// ---------------------------------------------------------------------
static inline void launch_gemm(const float* A, int lda, const float* W, int ldw,
                               const float* bias, float* Cmat, int ldc,
                               int M, int N, int K, hipStream_t stream)
{
    dim3 grid(M / 16, (N / 16 + 7) / 8);     // 8 N-tiles (one per wave) / block
    wmma_gemm_bias_f32<<<grid, 256, 0, stream>>>(A, lda, W, ldw, bias,
                                                 Cmat, ldc, M, N, K);
}

extern "C" void kernel_launch(void* const* d_in, const int* in_sizes, int n_in,
                              void* d_out, int out_size, void* d_ws, size_t ws_size,
                              hipStream_t stream)
{
    // setup_inputs() dict order
    const float* img   = (const float*)d_in[0];   // [B,T,D]
    const int*   label = (const int*)  d_in[1];   // [B,NSTEPS]
    const float* Wx    = (const float*)d_in[2];   // [A,D]
    const float* bx    = (const float*)d_in[3];   // [A]
    const float* Ws    = (const float*)d_in[4];   // [A,H]
    const float* bs    = (const float*)d_in[5];   // [A]
    const float* Ww    = (const float*)d_in[6];   // [1,A]
    const float* bw    = (const float*)d_in[7];   // [1]
    const float* emb   = (const float*)d_in[8];   // [C+1,A]
    const float* Wih   = (const float*)d_in[9];   // [3H, A+D]
    const float* bih   = (const float*)d_in[10];  // [3H]
    const float* Whh   = (const float*)d_in[11];  // [3H, H]
    const float* bhh   = (const float*)d_in[12];  // [3H]
    const float* Wfc   = (const float*)d_in[13];  // [C,H]
    const float* bfc   = (const float*)d_in[14];  // [C]
    float* out = (float*)d_out;                   // [B,NSTEPS,C]

    // ---- workspace carve (fp32), ~39.3 MB total
    char* ws = (char*)d_ws;
    float* xProj = (float*)ws; ws += (size_t)BB * TT * AA * sizeof(float);
    float* sProj = (float*)ws; ws += (size_t)BB * AA * sizeof(float);
    float* u     = (float*)ws; ws += (size_t)BB * (AA + DD) * sizeof(float);
    float* gi    = (float*)ws; ws += (size_t)BB * 3 * HH * sizeof(float);
    float* gh    = (float*)ws; ws += (size_t)BB * 3 * HH * sizeof(float);
    float* h     = (float*)ws; ws += (size_t)BB * HH * sizeof(float);

    // h0 = 0
    zero_f32<<<(BB * HH + 255) / 256, 256, 0, stream>>>(h, BB * HH);

    // xProj = img @ Wx^T + bx   (M=B*T=32768, N=A=256, K=D=512) — once
    launch_gemm(img, DD, Wx, DD, bx, xProj, AA, BB * TT, AA, DD, stream);

    for (int s = 0; s < NSTEPS; ++s) {
        // sProj = h @ Ws^T + bs            (512 x 256 x 256)
        launch_gemm(h, HH, Ws, HH, bs, sProj, AA, BB, AA, HH, stream);

        // attention: energies -> softmax -> ctx -> u = [emb[y] | ctx]
        attention_step<<<BB, 256, 0, stream>>>(xProj, sProj, img, Ww, bw,
                                               emb, label, u, s);

        // gi = u @ Wih^T + bih             (512 x 768 x 768)
        launch_gemm(u, AA + DD, Wih, AA + DD, bih, gi, 3 * HH, BB, 3 * HH, AA + DD, stream);
        // gh = h @ Whh^T + bhh             (512 x 768 x 256)
        launch_gemm(h, HH, Whh, HH, bhh, gh, 3 * HH, BB, 3 * HH, HH, stream);

        // GRU gates + hidden update
        gru_update<<<BB, 256, 0, stream>>>(gi, gh, h);

        // logits for this step: out[:, s, :] = h @ Wfc^T + bfc   (512 x 96 x 256)
        launch_gemm(h, HH, Wfc, HH, bfc, out + (size_t)s * CC, NSTEPS * CC,
                    BB, CC, HH, stream);
    }
}